// TokenBladeBank_8186207666894
// MI455X (gfx1250) — compile-verified
//
#include <hip/hip_runtime.h>
#include <cstdint>

// TokenBladeBank: FNV-1a-style hash of 4-grams -> gather 8 blade rows of 16 f32
// out[batch, seq, blade, d] = bank[blade, fnv(token_window[batch,seq,:]) % N_SLOTS, d]
//
// Pure gather / bandwidth problem (~70 MB total @ 23.3 TB/s ~ 3 us).
// wave32 mapping: 1 wave == 1 (batch,seq) position; lane = blade*4 + quad.
// Each lane: one 128-bit NT gather load from bank, one 128-bit NT store.

#define N_GRAM     4
#define D_STATE    16
#define N_SLOTS    500000u
#define N_BLADES   8
#define FNV_OFFSET 2166136261u
#define FNV_PRIME  16777619u

#define WAVES_PER_BLOCK 8
#define BLOCK (WAVES_PER_BLOCK * 32)   // 8 wave32 waves per workgroup

// clang-native 128-bit vector (HIP float4 is a class -> rejected by the builtins)
typedef float v4f __attribute__((ext_vector_type(4)));

__global__ __launch_bounds__(BLOCK)
void TokenBladeBank_gather_kernel(const unsigned long long* __restrict__ tok,  // int64 tokens, (n_pos, 4)
                                  const float* __restrict__ bank,              // (8, 500000, 16) f32
                                  float* __restrict__ out,                     // (n_pos, 8, 16) f32
                                  int n_pos)
{
    const int wave = threadIdx.x >> 5;        // wave32
    const int lane = threadIdx.x & 31;
    const int p    = blockIdx.x * WAVES_PER_BLOCK + wave;   // position = batch*8192 + seq
    if (p >= n_pos) return;

    // ---- FNV hash of the 4 tokens (int64 storage, values < 2^32) ----
    // All 32 lanes of the wave read the same 32B (one cacheline) -> L0 broadcast.
    const unsigned long long* tw = tok + (size_t)p * N_GRAM;

    // Prefetch the token stream a few blocks ahead (gfx1250 global_prefetch_b8).
    __builtin_prefetch(tw + (size_t)WAVES_PER_BLOCK * N_GRAM * 8, 0, 0);

    uint32_t h = FNV_OFFSET;
    h = (h ^ (uint32_t)tw[0]) * FNV_PRIME;
    h = (h ^ (uint32_t)tw[1]) * FNV_PRIME;
    h = (h ^ (uint32_t)tw[2]) * FNV_PRIME;
    h = (h ^ (uint32_t)tw[3]) * FNV_PRIME;
    const uint32_t addr = h % N_SLOTS;

    // ---- gather: 8 blades x 4 quads of float4 == 32 lanes (one wave32) ----
    const int blade = lane >> 2;   // 0..7
    const int quad  = lane & 3;    // 0..3  -> floats [quad*4, quad*4+4)

    const v4f* __restrict__ src =
        (const v4f*)(bank + ((size_t)blade * N_SLOTS + (size_t)addr) * D_STATE) + quad;
    // Bank rows are visited ~once each: non-temporal load keeps L2 for reuse elsewhere.
    v4f v = __builtin_nontemporal_load(src);

    // Fully coalesced 512B store per wave; write-once streaming -> non-temporal.
    v4f* __restrict__ dst =
        (v4f*)(out + (size_t)p * (N_BLADES * D_STATE)) + lane;   // lane == blade*4 + quad
    __builtin_nontemporal_store(v, dst);
}

extern "C" void kernel_launch(void* const* d_in, const int* in_sizes, int n_in,
                              void* d_out, int out_size, void* d_ws, size_t ws_size,
                              hipStream_t stream) {
    const unsigned long long* tok  = (const unsigned long long*)d_in[0]; // int64 token_window
    const float*              bank = (const float*)d_in[1];              // f32 bank
    float*                    out  = (float*)d_out;

    const int n_pos = in_sizes[0] / N_GRAM;                 // 8*8192 = 65536
    const int grid  = (n_pos + WAVES_PER_BLOCK - 1) / WAVES_PER_BLOCK;

    TokenBladeBank_gather_kernel<<<grid, BLOCK, 0, stream>>>(tok, bank, out, n_pos);
}